// L2Loss_38414187495817
// MI455X (gfx1250) — compile-verified
//
#include <hip/hip_runtime.h>
#include <hip/hip_bf16.h>
#include <math.h>

// Pairwise squared-distance loss:
//   pair_sum = N * sum(x^2) - dot(colsum, colsum);  out = exp(-pair_sum / (N*(N-1)/2))
//
// Memory-bound streaming reduction (64 MB, fits in 192 MB L2 on replays).
// ws layout (floats):
//   [0      .. 1023 ]  colsum accumulator (zeroed by k0, atomically built by k2)
//   [1024   .. 2047 ]  per-block sum-of-squares partials (G entries)
//   [2048   .. 2048+G*1024)  per-block column partial sums

#define DIM 1024

typedef float v2f __attribute__((ext_vector_type(2)));
typedef float v8f __attribute__((ext_vector_type(8)));

__global__ void l2_zero_kernel(float* __restrict__ colsum) {
    colsum[blockIdx.x * 256 + threadIdx.x] = 0.0f;   // 4 blocks x 256 = 1024
}

// Grid = G blocks x 256 threads. Block b handles rows b, b+G, b+2G, ...
// Thread t owns columns 4t..4t+3 (one float4 = whole row across the block).
__global__ void l2_stream_kernel(const float* __restrict__ feats,
                                 float* __restrict__ ws, int rows) {
    const int t = threadIdx.x;
    const float* base = feats + 4 * t;

    float4 colacc = make_float4(0.f, 0.f, 0.f, 0.f);
    float sq = 0.0f;

    const int stride = gridDim.x;
#pragma unroll 4
    for (int r = blockIdx.x; r < rows; r += stride) {
        const float4 v = *(const float4*)(base + (size_t)r * DIM);
        // prefetch same lane-offset a few rows ahead (speculative; OOB dropped)
        __builtin_prefetch(base + (size_t)(r + stride) * DIM, 0, 0);
        colacc.x += v.x; colacc.y += v.y; colacc.z += v.z; colacc.w += v.w;
        sq = fmaf(v.x, v.x, sq);
        sq = fmaf(v.y, v.y, sq);
        sq = fmaf(v.z, v.z, sq);
        sq = fmaf(v.w, v.w, sq);
    }

    // Column partials: every thread owns distinct columns -> plain coalesced store.
    float* part = ws + 2048 + (size_t)blockIdx.x * DIM;
    *(float4*)(part + 4 * t) = colacc;

    // Block-reduce sum of squares (wave32 shuffles + tiny LDS).
    __shared__ float red[8];
    float s = sq;
#pragma unroll
    for (int m = 16; m > 0; m >>= 1) s += __shfl_xor(s, m, 32);
    if ((t & 31) == 0) red[t >> 5] = s;
    __syncthreads();
    if (t == 0) {
        float tot = 0.0f;
#pragma unroll
        for (int w = 0; w < 8; ++w) tot += red[w];
        ws[1024 + blockIdx.x] = tot;
    }
}

// Grid = 64 blocks (4 column-groups x 16 p-groups) x 256 threads.
// Reduces ws_part[G][1024] into colsum[1024] with only 16K atomics total.
__global__ void l2_colreduce_kernel(const float* __restrict__ ws_part,
                                    float* __restrict__ colsum, int G) {
    const int cg = blockIdx.x & 3;
    const int pg = blockIdx.x >> 2;           // 0..15
    const int c  = cg * 256 + threadIdx.x;    // column
    float s = 0.0f;
    for (int p = pg; p < G; p += 16)
        s += ws_part[(size_t)p * DIM + c];
    atomicAdd(&colsum[c], s);
}

// One wave (32 threads): dot(colsum, colsum) = trace(C * C^T) via
// 16x V_WMMA_F32_16X16X4_F32. For a 16x4 f32 chunk, the A-operand layout and
// the transposed-chunk-as-B layout are element-identical, so the same v2f
// register pair feeds both A and B; diag of the accumulated D = dot product.
__global__ void l2_finalize_kernel(const float* __restrict__ ws,
                                   float* __restrict__ out, int rows, int G) {
    const int lane = threadIdx.x;             // 0..31
    const float* colsum = ws;                 // 1024 floats
    const float* sqp    = ws + 1024;          // G floats

    // Sum-of-squares partial pickup
    float sq = 0.0f;
    for (int i = lane; i < G; i += 32) sq += sqp[i];

    // WMMA dot over 16 chunks of 64 elements (chunk viewed as 16(m) x 4(k)).
    // Lane L<16 holds elem(m=L, k=0..1); lane L>=16 holds elem(m=L-16, k=2..3).
    const int m   = lane & 15;
    const int koff = (lane & 16) ? 2 : 0;
    v8f c = {};
#pragma unroll
    for (int q = 0; q < 16; ++q) {
        const v2f ab = *(const v2f*)(colsum + q * 64 + 4 * m + koff);
        c = __builtin_amdgcn_wmma_f32_16x16x4_f32(
                /*neg_a=*/false, ab, /*neg_b=*/false, ab,
                /*c_mod=*/(short)0, c, /*reuse_a=*/false, /*reuse_b=*/false);
    }

    // Extract trace: VGPR r holds M=r (lanes 0-15, N=lane) and M=r+8
    // (lanes 16-31, N=lane-16). Diagonal lives on lanes 0..7 and 24..31.
    float diag = 0.0f;
#pragma unroll
    for (int r = 0; r < 8; ++r) {
        if ((lane < 16 && lane == r) || (lane >= 16 && lane == r + 24))
            diag += c[r];
    }

    float dot = diag;
#pragma unroll
    for (int msk = 16; msk > 0; msk >>= 1) {
        dot += __shfl_xor(dot, msk, 32);
        sq  += __shfl_xor(sq,  msk, 32);
    }

    if (lane == 0) {
        const double n     = (double)rows;
        const double pair  = n * (double)sq - (double)dot;
        const double count = n * (n - 1.0) * 0.5;
        out[0] = (float)exp(-pair / count);
    }
}

extern "C" void kernel_launch(void* const* d_in, const int* in_sizes, int n_in,
                              void* d_out, int out_size, void* d_ws, size_t ws_size,
                              hipStream_t stream) {
    (void)n_in; (void)out_size;
    const float* feats = (const float*)d_in[0];
    float* ws  = (float*)d_ws;
    float* out = (float*)d_out;

    const int rows = in_sizes[0] / DIM;       // 16384 for the reference shapes

    // Size the partial-sum grid from available workspace (deterministic).
    long wsFloats = (long)(ws_size / sizeof(float));
    long avail = wsFloats - 2048;
    long g = avail / DIM;
    if (g > 1024) g = 1024;
    if (g > rows) g = rows;
    if (g < 1)    g = 1;
    const int G = (int)g;

    l2_zero_kernel<<<4, 256, 0, stream>>>(ws);
    l2_stream_kernel<<<G, 256, 0, stream>>>(feats, ws, rows);
    l2_colreduce_kernel<<<64, 256, 0, stream>>>(ws + 2048, ws, G);
    l2_finalize_kernel<<<1, 32, 0, stream>>>(ws, out, rows, G);
}